// GroupAttention_6133213299135
// MI455X (gfx1250) — compile-verified
//
#include <hip/hip_runtime.h>

// CDNA5 / gfx1250 fused windowed-attention kernel (fp32, wave32).
// One block = 128 threads = 4 waves, owns 8 (b,c,h_group) slabs of 8x32.
// Fully fused: x -> QKV(WMMA f32 16x16x4) -> attention(VALU) -> proj(WMMA)
// -> permuted store.  134 MB total HBM traffic (~5.8us floor @ 23.3 TB/s).

typedef __attribute__((ext_vector_type(2))) float v2f;
typedef __attribute__((ext_vector_type(8))) float v8f;

#define XR  36   // padded LDS row stride for 32-float x / y_pre rows
#define QR  100  // padded LDS row stride for 96-float qkv rows
#define WQR 104  // padded LDS row stride for w_qkv rows (96 cols)
#define WPR 40   // padded LDS row stride for w_proj rows (32 cols)

__global__ __launch_bounds__(128)
void ga_fused_kernel(const float* __restrict__ x,
                     const float* __restrict__ w_qkv,
                     const float* __restrict__ b_qkv,
                     const float* __restrict__ w_proj,
                     const float* __restrict__ b_proj,
                     float* __restrict__ out)
{
    __shared__ __align__(16) float s_xy[64 * XR];       // x staging, then y_pre rows
    __shared__ __align__(16) float s_qkv[8 * 8 * QR];   // qkv[u][s][f], then out staging
    __shared__ __align__(16) float s_wq[32 * WQR];      // w_qkv [k][f]
    __shared__ __align__(16) float s_wp[32 * WPR];      // w_proj [k][n]
    __shared__ float s_bq[96];
    __shared__ float s_bp[32];

    const int tid  = threadIdx.x;
    const int blk  = blockIdx.x;
    const int lane = tid & 31;
    const int wave = tid >> 5;   // 0..3
    const int half = lane >> 4;  // 0/1 (WMMA lane-half)
    const int lm   = lane & 15;

    // ---------------- Phase A: coalesced global -> LDS staging ----------------
    {
        const float4* xg = (const float4*)(x + (size_t)blk * 2048);
#pragma unroll
        for (int i2 = 0; i2 < 4; ++i2) {            // 512 float4 chunks of x
            int idx = i2 * 128 + tid;
            int row = idx >> 3, q4 = idx & 7;
            *(float4*)(s_xy + row * XR + q4 * 4) = xg[idx];
        }
        const float4* wg = (const float4*)w_qkv;    // 3072 floats = 768 chunks
#pragma unroll
        for (int i2 = 0; i2 < 6; ++i2) {
            int idx = i2 * 128 + tid;
            int row = idx / 24, q4 = idx % 24;      // 24 float4 per 96-float row
            *(float4*)(s_wq + row * WQR + q4 * 4) = wg[idx];
        }
        const float4* pg = (const float4*)w_proj;   // 1024 floats = 256 chunks
#pragma unroll
        for (int i2 = 0; i2 < 2; ++i2) {
            int idx = i2 * 128 + tid;
            int row = idx >> 3, q4 = idx & 7;
            *(float4*)(s_wp + row * WPR + q4 * 4) = pg[idx];
        }
        if (tid < 96)       s_bq[tid] = b_qkv[tid];
        else                s_bp[tid - 96] = b_proj[tid - 96];
    }
    __syncthreads();

    // ---------------- Phase B: QKV GEMM, D[16x96] = A[16x32] @ Wqkv + b -------
    // Wave w owns slabs 2w,2w+1 => 16 rows.  V_WMMA_F32_16X16X4_F32:
    //   A: lane<16 holds row lm cols {4kk,4kk+1}; lane>=16 cols {4kk+2,4kk+3}
    //   B: mirrored (v0 = K row 4kk+2*half, v1 = next K row), col = nt*16+lm
    //   C/D: vgpr i = row 8*half+i, col lm
    {
        const float* arow = s_xy + (wave * 16 + lm) * XR;
        v2f af[8];
#pragma unroll
        for (int kk = 0; kk < 8; ++kk) {
            af[kk][0] = arow[4 * kk + 2 * half];
            af[kk][1] = arow[4 * kk + 2 * half + 1];
        }
#pragma unroll
        for (int nt = 0; nt < 6; ++nt) {
            const int f = nt * 16 + lm;
            const float bias = s_bq[f];
            v8f acc = {bias, bias, bias, bias, bias, bias, bias, bias};
#pragma unroll
            for (int kk = 0; kk < 8; ++kk) {
                const int kr = 4 * kk + 2 * half;
                v2f bf;
                bf[0] = s_wq[kr * WQR + f];
                bf[1] = s_wq[(kr + 1) * WQR + f];
                acc = __builtin_amdgcn_wmma_f32_16x16x4_f32(
                          false, af[kk], false, bf, (short)0, acc, false, false);
            }
            // scatter: row m = 8*half+i  ->  slab u = 2*wave+half, s = i
            float* dst = s_qkv + (wave * 2 + half) * (8 * QR) + f;
#pragma unroll
            for (int i = 0; i < 8; ++i) dst[i * QR] = acc[i];
        }
    }
    __syncthreads();

    // ---------------- Phase C: attention, one (slab u, window gs, head n)/thread
    {
        const int u  = tid >> 4;        // 0..7
        const int gs = (tid >> 2) & 3;  // window in slab
        const int n  = tid & 3;         // head
        float q[8][2], k[8][2], vv[8][2];
#pragma unroll
        for (int t = 0; t < 8; ++t) {
            int s  = 2 * gs + (t >> 2);
            int fb = 24 * (t & 3) + 2 * n;
            const float* p = s_qkv + (u * 8 + s) * QR + fb;
            q[t][0]  = p[0];  q[t][1]  = p[1];
            k[t][0]  = p[8];  k[t][1]  = p[9];
            vv[t][0] = p[16]; vv[t][1] = p[17];
        }
        const float scale = 0.35355339059327373f;  // (H/nh)^-0.5 = 8^-0.5
        float o[16];
#pragma unroll
        for (int i = 0; i < 8; ++i) {
            float pr[8];
            float mx = -3.402823466e38f;
#pragma unroll
            for (int j = 0; j < 8; ++j) {
                pr[j] = (q[i][0] * k[j][0] + q[i][1] * k[j][1]) * scale;
                mx = fmaxf(mx, pr[j]);
            }
            float sum = 0.f, o0 = 0.f, o1 = 0.f;
#pragma unroll
            for (int j = 0; j < 8; ++j) {
                float e = __expf(pr[j] - mx);
                sum += e;
                o0 += e * vv[j][0];
                o1 += e * vv[j][1];
            }
            float inv = 1.0f / sum;
            o[2 * i]     = o0 * inv;
            o[2 * i + 1] = o1 * inv;
        }
        // pre-proj row r = 2n + gs/2, columns w = (gs&1)*16 + 2t + d
        float* yrow = s_xy + (u * 8 + 2 * n + (gs >> 1)) * XR + (gs & 1) * 16;
#pragma unroll
        for (int j = 0; j < 4; ++j)
            ((float4*)yrow)[j] =
                make_float4(o[4 * j], o[4 * j + 1], o[4 * j + 2], o[4 * j + 3]);
    }
    __syncthreads();

    // ---------------- Phase D: proj GEMM, D[16x32] = Y[16x32] @ Wproj + b -----
    {
        const float* arow = s_xy + (wave * 16 + lm) * XR;
        v2f af[8];
#pragma unroll
        for (int kk = 0; kk < 8; ++kk) {
            af[kk][0] = arow[4 * kk + 2 * half];
            af[kk][1] = arow[4 * kk + 2 * half + 1];
        }
#pragma unroll
        for (int nt = 0; nt < 2; ++nt) {
            const int col = nt * 16 + lm;
            const float bias = s_bp[col];
            v8f acc = {bias, bias, bias, bias, bias, bias, bias, bias};
#pragma unroll
            for (int kk = 0; kk < 8; ++kk) {
                const int kr = 4 * kk + 2 * half;
                v2f bf;
                bf[0] = s_wp[kr * WPR + col];
                bf[1] = s_wp[(kr + 1) * WPR + col];
                acc = __builtin_amdgcn_wmma_f32_16x16x4_f32(
                          false, af[kk], false, bf, (short)0, acc, false, false);
            }
            // stage result rows (reuse s_qkv as flat 64x32 output buffer)
            float* dst = s_qkv + (wave * 16 + 8 * half) * 32 + col;
#pragma unroll
            for (int i = 0; i < 8; ++i) dst[i * 32] = acc[i];
        }
    }
    __syncthreads();

    // ---------------- Phase E: permuted coalesced store ----------------------
    // row (u,r): slab uu=blk*8+u -> b=uu>>10, c=(uu>>2)&255, hg=uu&3;
    // n=r>>1, gh=r&1; h = n*8 + hg*2 + gh;
    // c3 = bit-permutation {c[7:6], c[3:1], c[5:4], c[0]}
    {
        const int uu0 = blk * 8;
        float4* og = (float4*)out;
#pragma unroll
        for (int i2 = 0; i2 < 4; ++i2) {
            int idx = i2 * 128 + tid;          // 512 float4 chunks
            int row = idx >> 3, q4 = idx & 7;  // 64 rows x 8 chunks
            int u = row >> 3, r = row & 7;
            int uu = uu0 + u;
            int bb = uu >> 10;
            int c  = (uu >> 2) & 255;
            int hg = uu & 3;
            int c3 = (c & 0xC0) | ((c & 0x0E) << 2) | ((c >> 3) & 0x06) | (c & 1);
            int h  = (r >> 1) * 8 + hg * 2 + (r & 1);
            int dst = ((bb * 256 + c3) * 32 + h) * 8 + q4;  // float4 units
            og[dst] = *(const float4*)(s_qkv + row * 32 + q4 * 4);
        }
    }
}

extern "C" void kernel_launch(void* const* d_in, const int* in_sizes, int n_in,
                              void* d_out, int out_size, void* d_ws, size_t ws_size,
                              hipStream_t stream) {
    (void)n_in; (void)out_size; (void)d_ws; (void)ws_size;
    const float* x      = (const float*)d_in[0];
    const float* w_qkv  = (const float*)d_in[1];
    const float* b_qkv  = (const float*)d_in[2];
    const float* w_proj = (const float*)d_in[3];
    const float* b_proj = (const float*)d_in[4];
    float* out = (float*)d_out;

    // in_sizes[0] = B*C*H*W; each block consumes 2048 input floats (8 slabs).
    const int blocks = in_sizes[0] / 2048;  // 8192 for B=64,C=256,H=W=32
    ga_fused_kernel<<<blocks, 128, 0, stream>>>(x, w_qkv, b_qkv, w_proj, b_proj, out);
}